// Correlation_33904471834986
// MI455X (gfx1250) — compile-verified
//
#include <hip/hip_runtime.h>

// Correlation layer (max_displacement=4) via banded-Gram WMMA f32 16x16x4,
// with double-buffered async global->LDS staging (ASYNCcnt path).
// out[b, dy*9+dx, h, w] = sum_c in1[b,c,h,w] * in2_pad[b,c,h+dy, w+dx]

#define BATCH 8
#define CH    256
#define HH    128
#define WW    128
#define DISP  4
#define ND    9              // 2*DISP+1
#define OUTD  81             // ND*ND
#define HB    8              // output rows per workgroup (one per wave)
#define WT    16             // output cols per workgroup (WMMA M/N)
#define CK    16             // channel chunk staged in LDS
#define R2    (HB + 2*DISP)  // 16 in2 rows staged
#define C2    (WT + 16)      // 32 in2 cols staged (two 16-wide B tiles)

// Channel-planar LDS layouts; plane strides padded so +2 planes == 16 dwords
// (mod 64 banks) -> conflict-free fragment reads, constant-offset ds loads.
// All strides are multiples of 4 dwords -> 16B-aligned b128 async targets.
#define P1    (HB*WT + 8)    // 136 dwords per in1 channel plane
#define P2    (R2*C2 + 8)    // 520 dwords per in2 channel plane

#define IN1_OFF 0
#define IN2_OFF (CK * P1)                    // 2176 floats
#define BUFSZ   (CK*P1 + CK*P2)              // 10496 floats per staging buffer
#define SMEM_FLOATS (2 * BUFSZ)              // double buffered; sout (10368) fits

typedef float v2f __attribute__((ext_vector_type(2)));
typedef float v8f __attribute__((ext_vector_type(8)));

__device__ __forceinline__ unsigned lds_addr(float* p) {
  return (unsigned)(uintptr_t)(__attribute__((address_space(3))) float*)p;
}

// Async DMA global->LDS, GVS addressing: SADDR(u64) + VADDR(i32) + ioffset.
__device__ __forceinline__ void async_b128(const float* base, int boff, float* lds) {
  asm volatile("global_load_async_to_lds_b128 %0, %1, %2"
               :: "v"(lds_addr(lds)), "v"(boff), "s"(base) : "memory");
}
__device__ __forceinline__ void async_b32(const float* base, int boff, float* lds) {
  asm volatile("global_load_async_to_lds_b32 %0, %1, %2"
               :: "v"(lds_addr(lds)), "v"(boff), "s"(base) : "memory");
}
__device__ __forceinline__ void wait_async0() {
  asm volatile("s_wait_asynccnt 0x0" ::: "memory");
}

__global__ __launch_bounds__(256, 1)
void corr_wmma_f32(const float* __restrict__ in1,
                   const float* __restrict__ in2,
                   float* __restrict__ out) {
  __shared__ float smem[SMEM_FLOATS];

  const int b    = blockIdx.z;
  const int h0   = blockIdx.y * HB;
  const int w0   = blockIdx.x * WT;
  const int tid  = threadIdx.x;
  const int lane = tid & 31;
  const int wave = tid >> 5;            // 0..7: owns output row h0+wave

  // Issue this thread's async loads for channel chunk cc into buffer buf.
  auto stage = [&](int cc, int buf) {
    const int boff = buf * BUFSZ;
    // in1 chunk: planar [c][hv][w]; 2 x b128 per thread, fully in-bounds.
#pragma unroll
    for (int i = 0; i < 2; ++i) {
      const int g  = tid + i * 256;
      const int wg = (g & 3) * 4;
      const int hv = (g >> 2) & 7;
      const int c  = g >> 5;
      const int go = (((b * CH + cc + c) * HH + (h0 + hv)) * WW + (w0 + wg)) * 4;
      async_b128(in1, go, &smem[boff + IN1_OFF + c * P1 + hv * WT + wg]);
    }
    // in2 chunk: planar [c][r][x]; rows y=h0-4+r, cols x=w0-4+j, zero-padded.
#pragma unroll
    for (int i = 0; i < 8; ++i) {
      const int g  = tid + i * 256;
      const int xg = (g & 7) * 4;
      const int r  = (g >> 3) & 15;
      const int c  = g >> 7;
      const int y  = h0 - DISP + r;
      float* dst = &smem[boff + IN2_OFF + c * P2 + r * C2 + xg];
      if ((unsigned)y < (unsigned)HH) {
        const int xg0 = w0 - DISP + xg;
        const int ro  = ((b * CH + cc + c) * HH + y) * WW;
        if (xg0 >= 0 && xg0 + 3 < WW) {
          async_b128(in2, (ro + xg0) * 4, dst);
        } else {
#pragma unroll
          for (int j = 0; j < 4; ++j) {
            const int x1 = xg0 + j;
            if ((unsigned)x1 < (unsigned)WW) async_b32(in2, (ro + x1) * 4, dst + j);
            else                             dst[j] = 0.0f;
          }
        }
      } else {
        dst[0] = 0.0f; dst[1] = 0.0f; dst[2] = 0.0f; dst[3] = 0.0f;
      }
    }
  };

  // WMMA 16x16x4 f32 per-lane fragment indices (A and B share the pattern):
  //   A[m,k]: m = lane%16, k = kbase + 2*(lane/16) + {0,1}
  //   B[k,n]: n = lane%16, k = kbase + 2*(lane/16) + {0,1}
  const int m  = lane & 15;
  const int kh = (lane >> 4) << 1;      // 0 or 2
  const int a_base = IN1_OFF + kh * P1 + wave * WT + m;
  const int b_base = IN2_OFF + kh * P2 + wave * C2 + m;

  // 18 accumulator chains: 9 dy x 2 column tiles (B col offsets -4 and +12)
  v8f acc[ND][2];
#pragma unroll
  for (int dy = 0; dy < ND; ++dy)
#pragma unroll
    for (int t = 0; t < 2; ++t)
#pragma unroll
      for (int e = 0; e < 8; ++e) acc[dy][t][e] = 0.0f;

  stage(0, 0);   // prologue: chunk 0 -> buffer 0

  for (int c0 = 0, it = 0; c0 < CH; c0 += CK, ++it) {
    const int cur  = it & 1;
    const int boff = cur * BUFSZ;

    wait_async0();       // this wave's async loads into buf[cur] landed
    __syncthreads();     // all waves' loads landed; prior compute done

    if (c0 + CK < CH) stage(c0 + CK, cur ^ 1);   // overlap DMA with compute

    // ---- WMMA band-matmul over this channel chunk ----
#pragma unroll
    for (int ks = 0; ks < CK / 4; ++ks) {
      v2f A;
      A.x = smem[boff + a_base + ks * 4 * P1];
      A.y = smem[boff + a_base + ks * 4 * P1 + P1];
#pragma unroll
      for (int t = 0; t < 2; ++t) {
        // Batch all 9 B-fragment loads so ds_loads overlap, then 9 WMMAs.
        v2f Bf[ND];
#pragma unroll
        for (int dy = 0; dy < ND; ++dy) {
          Bf[dy].x = smem[boff + b_base + ks * 4 * P2 + dy * C2 + 16 * t];
          Bf[dy].y = smem[boff + b_base + ks * 4 * P2 + dy * C2 + 16 * t + P2];
        }
#pragma unroll
        for (int dy = 0; dy < ND; ++dy)
          acc[dy][t] = __builtin_amdgcn_wmma_f32_16x16x4_f32(
              false, A, false, Bf[dy], (short)0, acc[dy][t], false, false);
      }
    }
  }

  __syncthreads();   // staging LDS free; reuse as sout[hv][d][w]

  // ---- extract band diagonals from C/D fragments into LDS ----
  // element (M,N) of a tile lives at lane N + 16*(M>=8), VGPR M%8
  const int Mhi = (lane >> 4) * 8;
#pragma unroll
  for (int dy = 0; dy < ND; ++dy)
#pragma unroll
    for (int t = 0; t < 2; ++t)
#pragma unroll
      for (int r = 0; r < 8; ++r) {
        const int M  = r + Mhi;
        const int dx = (lane & 15) + 16 * t - M;   // displacement index 0..8
        if (dx >= 0 && dx <= 2 * DISP) {
          const int d = dy * ND + dx;
          smem[(wave * OUTD + d) * WT + M] = acc[dy][t][r];
        }
      }
  __syncthreads();

  // ---- coalesced writeback: contiguous 16-float runs per (d, h) ----
  for (int e = tid; e < HB * OUTD * WT; e += 256) {
    const int wl = e & 15;
    const int d  = (e >> 4) % OUTD;
    const int hv = e / (OUTD * WT);
    out[(((size_t)b * OUTD + d) * HH + (h0 + hv)) * WW + (w0 + wl)] = smem[e];
  }
}

extern "C" void kernel_launch(void* const* d_in, const int* in_sizes, int n_in,
                              void* d_out, int out_size, void* d_ws, size_t ws_size,
                              hipStream_t stream) {
  const float* in1 = (const float*)d_in[0];
  const float* in2 = (const float*)d_in[1];
  float* out = (float*)d_out;
  dim3 grid(WW / WT, HH / HB, BATCH);   // 8 x 16 x 8 = 1024 workgroups
  corr_wmma_f32<<<grid, dim3(256, 1, 1), 0, stream>>>(in1, in2, out);
}